// CustomLinear_20504173871678
// MI455X (gfx1250) — compile-verified
//
#include <hip/hip_runtime.h>
#include <stdint.h>

// Fused 12-stage learned-twiddle radix-2 butterfly network, N=4096, D=4096.
// One workgroup owns COLS=8 columns entirely in LDS (SoA re[]/im[], 256 KB of
// the WGP's 320 KB), loaded once from HBM via the Tensor Data Mover with the
// initial i^(N/2) permutation folded into the DMA destination, then 12 stages
// of butterflies in LDS, then a single interleaved-complex64 store.

#define N      4096
#define LOGN   12
#define COLS   8
#define THREADS 512
#define WAVES  (THREADS / 32)
#define ROWS_PER_WAVE (N / WAVES)   // 256

typedef unsigned int u32x4 __attribute__((ext_vector_type(4)));
typedef int          i32x8 __attribute__((ext_vector_type(8)));
typedef int          i32x4 __attribute__((ext_vector_type(4)));

__global__ __launch_bounds__(THREADS)
void fft_butterfly_fused(const float* __restrict__ x,
                         const float* __restrict__ wts,
                         float2* __restrict__ out) {
  extern __shared__ float smem[];
  float* re = smem;              // [N][COLS] packed f32 (TDM-friendly)
  float* im = smem + N * COLS;   // [N][COLS]

  const int tid  = threadIdx.x;
  const int wave = tid >> 5;
  const int c0   = blockIdx.x * COLS;

  // ---- TDM: each wave DMAs a 256-row x 8-col f32 tile global->LDS.
  // Initial permutation newx[i] = x[i ^ (N/2)] is applied by the choice of
  // LDS destination row (gr ^ 2048), exact since 256 | 2048.
  {
    const int gr = wave * ROWS_PER_WAVE;      // global row start
    const int lr = gr ^ (N / 2);              // LDS row start (permuted)
    const uint64_t ga = (uint64_t)(uintptr_t)(x + (size_t)gr * N + c0);
    // Generic LDS pointer: addr[31:0] IS the LDS byte offset (aperture spec).
    const uint32_t lds_off = (uint32_t)(uintptr_t)(re + (size_t)lr * COLS);

    u32x4 g0;
    g0[0] = 1u;                                        // count=1 (valid D#)
    g0[1] = lds_off;                                   // lds_addr (bytes)
    g0[2] = (uint32_t)(ga & 0xFFFFFFFFull);            // global_addr[31:0]
    g0[3] = (uint32_t)((ga >> 32) & 0x01FFFFFFull)     // global_addr[56:32]
          | (2u << 30);                                // type = 2 ("image")

    i32x8 g1;
    g1[0] = (int)(2u << 16);                           // data_size=2 -> 4 bytes
    g1[1] = (int)(((uint32_t)N & 0xFFFFu) << 16);      // tensor_dim0 lo16 @63:48
    g1[2] = (int)(((uint32_t)N & 0xFFFFu) << 16);      // tensor_dim1 lo16 @95:80
    g1[3] = (int)((uint32_t)COLS << 16);               // tile_dim0=8 @127:112
    g1[4] = (int)(uint32_t)ROWS_PER_WAVE;              // tile_dim1=256 @143:128
    g1[5] = (int)(uint32_t)N;                          // tensor_dim0_stride=4096
    g1[6] = 0;
    g1[7] = 0;

    i32x4 gz  = {0, 0, 0, 0};                          // 2-D tile: groups 2/3 unused
    i32x8 gz8 = {0, 0, 0, 0, 0, 0, 0, 0};              // extra group (clang-23 form)
    __builtin_amdgcn_tensor_load_to_lds(g0, g1, gz, gz, gz8, 0);
  }

  // Zero the imaginary plane while the DMA is in flight (disjoint LDS region).
  for (int t = tid; t < N * COLS; t += THREADS) im[t] = 0.0f;

  __builtin_amdgcn_s_wait_tensorcnt(0);
  __syncthreads();

  // ---- 12 radix-2 stages with learned twiddles.
  // Hardware v_sin/v_cos take input in revolutions (x * 2*pi), so feed
  // frac = -(k/N)*w directly: exp(-2*pi*i*(k/N)*w) = cos(2*pi*frac)+i*sin(...).
  const float inv_n = 1.0f / (float)N;
  for (int s = 1; s <= LOGN; ++s) {
    const int half = 1 << (s - 1);
    for (int t = tid; t < (N / 2) * COLS; t += THREADS) {
      const int c = t & (COLS - 1);
      const int b = t >> 3;                        // butterfly index in column
      const int r = b & (half - 1);
      const int j = ((b >> (s - 1)) << s) + r;     // top row
      const int k = r << (LOGN - s);               // twiddle row index
      const float wv   = wts[((size_t)k << LOGN) + c0 + c];
      const float frac = -(float)k * inv_n * wv;   // phase / (2*pi)
      const float sn = __builtin_amdgcn_sinf(frac);
      const float cs = __builtin_amdgcn_cosf(frac);
      const int ti = j * COLS + c;
      const int bi = (j + half) * COLS + c;
      const float br = re[bi], bmi = im[bi];
      const float tr  = cs * br  - sn * bmi;       // tw * bot
      const float tmi = cs * bmi + sn * br;
      const float ar = re[ti], ai = im[ti];
      re[ti] = ar + tr;   im[ti] = ai + tmi;
      re[bi] = ar - tr;   im[bi] = ai - tmi;
    }
    __syncthreads();
  }

  // ---- Store interleaved complex64 (matches numpy complex64 layout).
  for (int t = tid; t < N * COLS; t += THREADS) {
    const int c   = t & (COLS - 1);
    const int row = t >> 3;
    out[(size_t)row * N + c0 + c] = make_float2(re[t], im[t]);
  }
}

extern "C" void kernel_launch(void* const* d_in, const int* in_sizes, int n_in,
                              void* d_out, int out_size, void* d_ws, size_t ws_size,
                              hipStream_t stream) {
  const float* x = (const float*)d_in[0];
  const float* w = (const float*)d_in[1];
  float2* out = (float2*)d_out;
  const size_t shmem = (size_t)2 * N * COLS * sizeof(float);  // 256 KB / WGP
  fft_butterfly_fused<<<N / COLS, THREADS, shmem, stream>>>(x, w, out);
  (void)in_sizes; (void)n_in; (void)out_size; (void)d_ws; (void)ws_size;
}